// RefractiveSurface_71193377899036
// MI455X (gfx1250) — compile-verified
//
#include <hip/hip_runtime.h>

// ---------------------------------------------------------------------------
// RefractiveSurface: elementwise ray/parabola intersection + refraction.
// Memory-bound streaming kernel: 32 B per ray, ~134 MB total -> ~5.8 us floor
// at 23.3 TB/s. Strategy: 2 rays per lane => 128-bit nontemporal global
// loads/stores, wave32, max occupancy. Hardware rcp/rsq/sqrt (1-instruction,
// ~1 ulp) instead of precise div/sqrt expansions keep VALU far below the
// memory roofline. No matrix structure -> no WMMA.
// ---------------------------------------------------------------------------

typedef float fv4 __attribute__((ext_vector_type(4)));

#define A_COEF 0.01f
#define RATIO  (1.0f / 1.5f)   // N1/N2
#define EPS_Q  1e-10f
#define BIG_T  1e30f

__device__ __forceinline__ float fast_rcp(float x)  { return __builtin_amdgcn_rcpf(x); }   // v_rcp_f32
__device__ __forceinline__ float fast_rsq(float x)  { return __builtin_amdgcn_rsqf(x); }   // v_rsq_f32
__device__ __forceinline__ float fast_sqrt(float x) { return __builtin_amdgcn_sqrtf(x); }  // v_sqrt_f32

__device__ __forceinline__ void refract_one(float ox, float oy, float dx, float dy,
                                            float target,
                                            float& cx, float& cy,
                                            float& rx, float& ry)
{
    // Quadratic: A*dy^2 * t^2 + (2A*oy*dy - dx) * t + (A*oy^2 + target - ox) = 0
    float a_q = A_COEF * dy * dy;
    float b_q = 2.0f * A_COEF * oy * dy - dx;
    float c_q = A_COEF * oy * oy + target - ox;

    bool  lin    = fabsf(a_q) < EPS_Q;
    float a_safe = lin ? 1.0f : a_q;

    float disc = fmaxf(b_q * b_q - 4.0f * a_q * c_q, 0.0f);
    float sq   = fast_sqrt(disc);
    float sgn  = (b_q >= 0.0f) ? 1.0f : -1.0f;
    float q    = -0.5f * (b_q + sgn * sq);
    float q_safe = (fabsf(q) < EPS_Q) ? EPS_Q : q;

    float t1 = q * fast_rcp(a_safe);
    float t2 = c_q * fast_rcp(q_safe);
    float t_quad = fminf((t1 > 0.0f) ? t1 : BIG_T,
                         (t2 > 0.0f) ? t2 : BIG_T);

    float b_safe = (fabsf(b_q) < EPS_Q) ? EPS_Q : b_q;
    float t_lin  = -c_q * fast_rcp(b_safe);

    float t = lin ? t_lin : t_quad;

    // Collision point
    cx = ox + t * dx;
    cy = oy + t * dy;

    // Surface normal of x = A*y^2 + target:  n = normalize([1, -2A*y])
    float nyr  = -2.0f * A_COEF * cy;
    float ninv = fast_rsq(1.0f + nyr * nyr);
    float nx = ninv;
    float ny = nyr * ninv;

    // Orient normal against incoming ray
    float dot = nx * dx + ny * dy;
    if (dot > 0.0f) { nx = -nx; ny = -ny; }

    // Normalize ray direction
    float vinv = fast_rsq(dx * dx + dy * dy);
    float vx = dx * vinv;
    float vy = dy * vinv;

    // Snell refraction
    float cosi = -(nx * vx + ny * vy);
    float k    = fmaxf(1.0f - RATIO * RATIO * (1.0f - cosi * cosi), 0.0f);
    float coef = RATIO * cosi - fast_sqrt(k);

    rx = RATIO * vx + coef * nx;
    ry = RATIO * vy + coef * ny;
}

// Vector path: one float4 = 2 rays per lane. Requires even ray count.
__global__ __launch_bounds__(256) void refract_kernel_v4(
    const float* __restrict__ org,
    const float* __restrict__ vec,
    const float* __restrict__ target_p,
    float* __restrict__ coll,
    float* __restrict__ refr,
    float* __restrict__ tgt_out,
    int npairs)
{
    int i = blockIdx.x * blockDim.x + threadIdx.x;
    float target = *target_p;          // uniform scalar
    if (i == 0) *tgt_out = target;     // pass-through third output
    if (i >= npairs) return;

    const fv4* o4 = (const fv4*)org;
    const fv4* v4 = (const fv4*)vec;
    fv4* c4 = (fv4*)coll;
    fv4* r4 = (fv4*)refr;

    fv4 o = __builtin_nontemporal_load(&o4[i]);   // global_load_b128 th:NT
    fv4 d = __builtin_nontemporal_load(&v4[i]);

    float c0x, c0y, r0x, r0y;
    float c1x, c1y, r1x, r1y;
    refract_one(o.x, o.y, d.x, d.y, target, c0x, c0y, r0x, r0y);
    refract_one(o.z, o.w, d.z, d.w, target, c1x, c1y, r1x, r1y);

    fv4 c = { c0x, c0y, c1x, c1y };
    fv4 r = { r0x, r0y, r1x, r1y };

    __builtin_nontemporal_store(c, &c4[i]);       // global_store_b128 th:NT
    __builtin_nontemporal_store(r, &r4[i]);
}

// Scalar fallback (odd ray counts; keeps every access naturally aligned)
__global__ __launch_bounds__(256) void refract_kernel_s(
    const float* __restrict__ org,
    const float* __restrict__ vec,
    const float* __restrict__ target_p,
    float* __restrict__ coll,
    float* __restrict__ refr,
    float* __restrict__ tgt_out,
    int nrays)
{
    int i = blockIdx.x * blockDim.x + threadIdx.x;
    float target = *target_p;
    if (i == 0) *tgt_out = target;
    if (i >= nrays) return;

    float ox = org[2 * i + 0];
    float oy = org[2 * i + 1];
    float dx = vec[2 * i + 0];
    float dy = vec[2 * i + 1];

    float cx, cy, rx, ry;
    refract_one(ox, oy, dx, dy, target, cx, cy, rx, ry);

    coll[2 * i + 0] = cx;
    coll[2 * i + 1] = cy;
    refr[2 * i + 0] = rx;
    refr[2 * i + 1] = ry;
}

extern "C" void kernel_launch(void* const* d_in, const int* in_sizes, int n_in,
                              void* d_out, int out_size, void* d_ws, size_t ws_size,
                              hipStream_t stream)
{
    const float* org = (const float*)d_in[0];   // rays_origins (N,2) f32
    const float* vec = (const float*)d_in[1];   // rays_vectors (N,2) f32
    const float* tgt = (const float*)d_in[2];   // target scalar f32

    const int nrays = in_sizes[0] / 2;

    float* out      = (float*)d_out;
    float* coll     = out;                              // N*2 floats
    float* refr     = out + (size_t)2 * (size_t)nrays;  // N*2 floats
    float* tgt_out  = out + (size_t)4 * (size_t)nrays;  // 1 float

    const int block = 256;

    if ((nrays & 1) == 0) {
        int npairs = nrays >> 1;
        int nthreads = npairs > 0 ? npairs : 1;  // still write target scalar
        int grid = (nthreads + block - 1) / block;
        refract_kernel_v4<<<grid, block, 0, stream>>>(org, vec, tgt,
                                                      coll, refr, tgt_out, npairs);
    } else {
        int nthreads = nrays > 0 ? nrays : 1;
        int grid = (nthreads + block - 1) / block;
        refract_kernel_s<<<grid, block, 0, stream>>>(org, vec, tgt,
                                                     coll, refr, tgt_out, nrays);
    }
}